// RGConv_14448269984554
// MI455X (gfx1250) — compile-verified
//
#include <hip/hip_runtime.h>
#include <hip/hip_bf16.h>

// ---------------------------------------------------------------------------
// RGConv for MI455X (gfx1250, wave32, WMMA).
// All GEMMs use V_WMMA_F32_16X16X4_F32 (full fp32 matrix op) because the
// KNN top-k ranking is precision-sensitive and the whole problem (~22 GFLOP,
// ~40MB traffic) is tiny vs 23.3 TB/s HBM — fp32 WMMA is the right path.
// BN reductions use fixed-slot partials + fixed-order tree (deterministic).
// ---------------------------------------------------------------------------

static constexpr int BB = 4;
static constexpr int NN = 4096;
static constexpr int DD = 128;
static constexpr int HH = 64;
static constexpr int KK = 16;
static constexpr int OO = 128;
static constexpr int BN = BB * NN;          // 16384 rows
static constexpr int NE = BN * KK;          // 262144 edge rows

typedef __attribute__((ext_vector_type(2))) float v2f;
typedef __attribute__((ext_vector_type(4))) float v4f;
typedef __attribute__((ext_vector_type(8))) float v8f;

__device__ __forceinline__ v2f ld2(const float* p) {
    return *reinterpret_cast<const v2f*>(p);
}

// D = A(16x4 f32) * B(4x16 f32) + C(16x16 f32).  8-arg pattern per probe:
// (neg_a, A, neg_b, B, c_mod, C, reuse_a, reuse_b)
__device__ __forceinline__ v8f wmma4(v2f a, v2f b, v8f c) {
    return __builtin_amdgcn_wmma_f32_16x16x4_f32(false, a, false, b,
                                                 (short)0, c, false, false);
}

__device__ __forceinline__ float gelu_exact(float h) {
    return 0.5f * h * (1.0f + erff(h * 0.70710678118654752440f));
}

// ---------------------------------------------------------------------------
// 1) xt = x @ ft_w + ft_b   (M=16384, N=128, K=128) — one wave per 16x16 tile
// ---------------------------------------------------------------------------
__global__ void __launch_bounds__(32)
ft_gemm_kernel(const float* __restrict__ x, const float* __restrict__ ftw,
               const float* __restrict__ ftb, float* __restrict__ xt) {
    const int lane = threadIdx.x;
    const int mt = blockIdx.x >> 3;      // 1024 row tiles
    const int ct = blockIdx.x & 7;       // 8 col tiles
    const int col  = lane & 15;
    const int koff = (lane >> 4) * 2;
    const int ncol = ct * 16 + col;
    const float* xr = x + (size_t)(mt * 16 + col) * DD + koff;
    v8f c = {};
#pragma unroll
    for (int kk = 0; kk < 32; ++kk) {
        const int d0 = kk * 4 + koff;
        v2f a  = ld2(xr + kk * 4);
        v2f bf = { ftw[d0 * DD + ncol], ftw[(d0 + 1) * DD + ncol] };
        c = wmma4(a, bf, c);
    }
    const float bias = ftb[ncol];
#pragma unroll
    for (int r = 0; r < 8; ++r) {
        const int M = mt * 16 + ((lane < 16) ? r : (8 + r));
        xt[(size_t)M * DD + ncol] = c[r] + bias;
    }
}

// ---------------------------------------------------------------------------
// 2) xx[row] = sum(xt[row]^2)
// ---------------------------------------------------------------------------
__global__ void rownorm_kernel(const float* __restrict__ xt, float* __restrict__ xx) {
    const int row = blockIdx.x * 256 + threadIdx.x;  // grid 64
    const v4f* p = reinterpret_cast<const v4f*>(xt + (size_t)row * DD);
    float s = 0.0f;
#pragma unroll
    for (int i = 0; i < 32; ++i) {
        v4f v = p[i];
        s += v[0] * v[0] + v[1] * v[1] + v[2] * v[2] + v[3] * v[3];
    }
    xx[row] = s;
}

// ---------------------------------------------------------------------------
// 3) Fused distance GEMM + per-row top-17 (smallest) selection.
//    One wave per (batch, 16-row tile). A-fragments hoisted (64 VGPRs),
//    reused across all 256 column tiles.  dist' = xx_m - 2*dot (row term is
//    a per-row constant shift -> identical ranking; self = -xx_n = minimum).
// ---------------------------------------------------------------------------
__global__ void __launch_bounds__(32)
dist_topk_kernel(const float* __restrict__ xt, const float* __restrict__ xx,
                 int* __restrict__ nbr) {
    const int lane = threadIdx.x;
    const int b  = blockIdx.x >> 8;       // 256 tiles per batch
    const int mt = blockIdx.x & 255;
    const int row0 = mt * 16;
    const int col  = lane & 15;
    const int koff = (lane >> 4) * 2;
    const float* xtb = xt + (size_t)b * NN * DD;
    const float* xxb = xx + b * NN;

    v2f af[32];
    {
        const float* ar = xtb + (size_t)(row0 + col) * DD + koff;
#pragma unroll
        for (int kk = 0; kk < 32; ++kk) af[kk] = ld2(ar + kk * 4);
    }

    float bd[KK + 1];
    int   bi[KK + 1];
#pragma unroll
    for (int j = 0; j <= KK; ++j) { bd[j] = 3.0e38f; bi[j] = 0; }

    __shared__ float tile[16][17];

    for (int t = 0; t < 256; ++t) {
        const int m0 = t * 16;
        const float* br = xtb + (size_t)(m0 + col) * DD + koff;
        if (t < 255) __builtin_prefetch(br + 16 * DD, 0, 0);
        v8f c = {};
#pragma unroll
        for (int kk = 0; kk < 32; ++kk) {
            v2f bf = ld2(br + kk * 4);
            c = wmma4(af[kk], bf, c);
        }
        const float xxc = xxb[m0 + col];
#pragma unroll
        for (int r = 0; r < 8; ++r) {
            const int M = (lane < 16) ? r : (8 + r);
            tile[M][col] = xxc - 2.0f * c[r];
        }
        __syncthreads();
        if (lane < 16) {
#pragma unroll
            for (int j = 0; j < 16; ++j) {
                const float cv = tile[lane][j];
                const int   ci = m0 + j;
                if (cv < bd[KK]) {            // strict < : earlier index wins ties
                    bd[KK] = cv; bi[KK] = ci;
#pragma unroll
                    for (int jj = KK; jj > 0; --jj) {
                        if (bd[jj] < bd[jj - 1]) {
                            float td = bd[jj]; bd[jj] = bd[jj - 1]; bd[jj - 1] = td;
                            int   ti = bi[jj]; bi[jj] = bi[jj - 1]; bi[jj - 1] = ti;
                        }
                    }
                }
            }
        }
        __syncthreads();
    }
    if (lane < 16) {
        int* out = nbr + ((size_t)b * NN + row0 + lane) * KK;
#pragma unroll
        for (int k = 0; k < KK; ++k) out[k] = bi[k + 1];  // drop self (slot 0)
    }
}

// ---------------------------------------------------------------------------
// 4) Edge MLP pass 1: hpre = edge @ rw1_w + rw1_b ; deterministic per-wave
//    channel sum / sumsq partials (16384 waves x 64 channels).
// ---------------------------------------------------------------------------
__global__ void __launch_bounds__(32)
edge_stats_kernel(const float* __restrict__ x, const int* __restrict__ nbr,
                  const float* __restrict__ rw1w, const float* __restrict__ rw1b,
                  float* __restrict__ p1s, float* __restrict__ p1q) {
    const int lane = threadIdx.x;
    const int wid  = blockIdx.x;          // 16384 tiles of 16 edge rows
    const int row0 = wid * 16;
    const int col  = lane & 15;
    const int koff = (lane >> 4) * 2;
    const int e  = row0 + col;
    const int bn = e >> 4;                // / K
    const int bb = bn >> 12;              // / N
    const int ni = nbr[e];
    const float* xs = x + (size_t)bn * DD;
    const float* xn = x + (size_t)((bb << 12) + ni) * DD;

    v2f af[32];
#pragma unroll
    for (int kk = 0; kk < 32; ++kk) {
        const int d0 = kk * 4 + koff;
        af[kk] = ld2(xn + d0) - ld2(xs + d0);
    }
#pragma unroll
    for (int nt = 0; nt < 4; ++nt) {
        const int ch = nt * 16 + col;
        v8f c = {};
#pragma unroll
        for (int kk = 0; kk < 32; ++kk) {
            const int d0 = kk * 4 + koff;
            v2f bf = { rw1w[d0 * HH + ch], rw1w[(d0 + 1) * HH + ch] };
            c = wmma4(af[kk], bf, c);
        }
        const float bias = rw1b[ch];
        float s = 0.0f, q = 0.0f;
#pragma unroll
        for (int r = 0; r < 8; ++r) { const float h = c[r] + bias; s += h; q += h * h; }
        s += __shfl_down(s, 16);
        q += __shfl_down(q, 16);
        if (lane < 16) {
            p1s[(size_t)ch * 16384 + wid] = s;
            p1q[(size_t)ch * 16384 + wid] = q;
        }
    }
}

// ---------------------------------------------------------------------------
// 5) BN1 finalize: fold mean/var + rw1 bias into scale/shift per channel.
// ---------------------------------------------------------------------------
__global__ void bn1_fin_kernel(const float* __restrict__ p1s, const float* __restrict__ p1q,
                               const float* __restrict__ g, const float* __restrict__ beta,
                               const float* __restrict__ rw1b,
                               float* __restrict__ scale1, float* __restrict__ shift1) {
    __shared__ float ls[256], lq[256];
    const int c = blockIdx.x;             // 64
    const int t = threadIdx.x;            // 256
    float s = 0.0f, q = 0.0f;
    for (int w = t; w < 16384; w += 256) {
        s += p1s[(size_t)c * 16384 + w];
        q += p1q[(size_t)c * 16384 + w];
    }
    ls[t] = s; lq[t] = q; __syncthreads();
    for (int off = 128; off > 0; off >>= 1) {
        if (t < off) { ls[t] += ls[t + off]; lq[t] += lq[t + off]; }
        __syncthreads();
    }
    if (t == 0) {
        const float m  = ls[0] * (1.0f / (float)NE);
        const float v  = lq[0] * (1.0f / (float)NE) - m * m;
        const float sc = g[c] * rsqrtf(v + 1e-5f);
        scale1[c] = sc;
        shift1[c] = (rw1b[c] - m) * sc + beta[c];   // bias folded into shift
    }
}

// ---------------------------------------------------------------------------
// 6) Edge MLP pass 2: recompute, apply BN + exact GELU, matvec rw2, sigmoid.
// ---------------------------------------------------------------------------
__global__ void __launch_bounds__(32)
edge_weight_kernel(const float* __restrict__ x, const int* __restrict__ nbr,
                   const float* __restrict__ rw1w,
                   const float* __restrict__ scale1, const float* __restrict__ shift1,
                   const float* __restrict__ rw2w, const float* __restrict__ rw2b,
                   float* __restrict__ wgt) {
    const int lane = threadIdx.x;
    const int wid  = blockIdx.x;
    const int row0 = wid * 16;
    const int col  = lane & 15;
    const int koff = (lane >> 4) * 2;
    const int e  = row0 + col;
    const int bn = e >> 4;
    const int bb = bn >> 12;
    const int ni = nbr[e];
    const float* xs = x + (size_t)bn * DD;
    const float* xn = x + (size_t)((bb << 12) + ni) * DD;

    v2f af[32];
#pragma unroll
    for (int kk = 0; kk < 32; ++kk) {
        const int d0 = kk * 4 + koff;
        af[kk] = ld2(xn + d0) - ld2(xs + d0);
    }
    __shared__ float hl[16][65];
#pragma unroll
    for (int nt = 0; nt < 4; ++nt) {
        const int ch = nt * 16 + col;
        v8f c = {};
#pragma unroll
        for (int kk = 0; kk < 32; ++kk) {
            const int d0 = kk * 4 + koff;
            v2f bf = { rw1w[d0 * HH + ch], rw1w[(d0 + 1) * HH + ch] };
            c = wmma4(af[kk], bf, c);
        }
        const float sc = scale1[ch], sh = shift1[ch];
#pragma unroll
        for (int r = 0; r < 8; ++r) {
            float h = c[r] * sc + sh;
            h = gelu_exact(h);
            const int M = (lane < 16) ? r : (8 + r);
            hl[M][ch] = h;
        }
    }
    __syncthreads();
    if (lane < 16) {
        float acc = rw2b[0];
#pragma unroll
        for (int cc = 0; cc < HH; ++cc) acc += hl[lane][cc] * rw2w[cc];
        wgt[row0 + lane] = 1.0f / (1.0f + expf(-acc));
    }
}

// ---------------------------------------------------------------------------
// 7) agg[b,n,:] = sum_k w * (x[nbr] - x[n])
// ---------------------------------------------------------------------------
__global__ void aggregate_kernel(const float* __restrict__ x, const int* __restrict__ nbr,
                                 const float* __restrict__ wgt, float* __restrict__ agg) {
    const int bn = blockIdx.x;            // 16384
    const int d  = threadIdx.x;           // 128
    const int bb = bn >> 12;
    const float xv = x[(size_t)bn * DD + d];
    const int base = bn * KK;
    float acc = 0.0f;
#pragma unroll 4
    for (int k = 0; k < KK; ++k) {
        const int   nb = nbr[base + k];
        const float wk = wgt[base + k];
        acc = fmaf(wk, x[(size_t)((bb << 12) + nb) * DD + d] - xv, acc);
    }
    agg[(size_t)bn * DD + d] = acc;
}

// ---------------------------------------------------------------------------
// 8) upre = [x | agg] @ nu_w + nu_b (M=16384, N=128, K=256) + stats partials.
// ---------------------------------------------------------------------------
__global__ void __launch_bounds__(32)
nu_gemm_kernel(const float* __restrict__ x, const float* __restrict__ agg,
               const float* __restrict__ nuw, const float* __restrict__ nub,
               float* __restrict__ upre, float* __restrict__ p2s, float* __restrict__ p2q) {
    const int lane = threadIdx.x;
    const int mt = blockIdx.x >> 3;       // 1024 row tiles
    const int ct = blockIdx.x & 7;        // 8 col tiles
    const int col  = lane & 15;
    const int koff = (lane >> 4) * 2;
    const int ch = ct * 16 + col;
    const float* xr = x   + (size_t)(mt * 16 + col) * DD;
    const float* ar = agg + (size_t)(mt * 16 + col) * DD;
    v8f c = {};
#pragma unroll
    for (int kk = 0; kk < 64; ++kk) {
        const int d0 = kk * 4 + koff;
        const float* asrc = (kk < 32) ? (xr + d0) : (ar + d0 - DD);
        v2f a  = ld2(asrc);
        v2f bf = { nuw[d0 * OO + ch], nuw[(d0 + 1) * OO + ch] };
        c = wmma4(a, bf, c);
    }
    const float bias = nub[ch];
    float s = 0.0f, q = 0.0f;
#pragma unroll
    for (int r = 0; r < 8; ++r) {
        const int M = mt * 16 + ((lane < 16) ? r : (8 + r));
        const float u = c[r] + bias;
        upre[(size_t)M * OO + ch] = u;
        s += u; q += u * u;
    }
    s += __shfl_down(s, 16);
    q += __shfl_down(q, 16);
    if (lane < 16) {
        p2s[(size_t)ch * 1024 + mt] = s;
        p2q[(size_t)ch * 1024 + mt] = q;
    }
}

// ---------------------------------------------------------------------------
// 9) BN2 finalize (bias already inside upre).
// ---------------------------------------------------------------------------
__global__ void bn2_fin_kernel(const float* __restrict__ p2s, const float* __restrict__ p2q,
                               const float* __restrict__ g, const float* __restrict__ beta,
                               float* __restrict__ scale2, float* __restrict__ shift2) {
    __shared__ float ls[256], lq[256];
    const int c = blockIdx.x;             // 128
    const int t = threadIdx.x;            // 256
    float s = 0.0f, q = 0.0f;
    for (int w = t; w < 1024; w += 256) {
        s += p2s[(size_t)c * 1024 + w];
        q += p2q[(size_t)c * 1024 + w];
    }
    ls[t] = s; lq[t] = q; __syncthreads();
    for (int off = 128; off > 0; off >>= 1) {
        if (t < off) { ls[t] += ls[t + off]; lq[t] += lq[t + off]; }
        __syncthreads();
    }
    if (t == 0) {
        const float m  = ls[0] * (1.0f / (float)BN);
        const float v  = lq[0] * (1.0f / (float)BN) - m * m;
        const float sc = g[c] * rsqrtf(v + 1e-5f);
        scale2[c] = sc;
        shift2[c] = beta[c] - m * sc;
    }
}

// ---------------------------------------------------------------------------
// 10) out = gelu(upre * scale2 + shift2)
// ---------------------------------------------------------------------------
__global__ void out_kernel(const float* __restrict__ upre,
                           const float* __restrict__ scale2,
                           const float* __restrict__ shift2,
                           float* __restrict__ out) {
    const int i  = blockIdx.x * 256 + threadIdx.x;   // grid 8192
    const int ch = i & 127;
    const float u = upre[i] * scale2[ch] + shift2[ch];
    out[i] = gelu_exact(u);
}

// ---------------------------------------------------------------------------
extern "C" void kernel_launch(void* const* d_in, const int* in_sizes, int n_in,
                              void* d_out, int out_size, void* d_ws, size_t ws_size,
                              hipStream_t stream) {
    const float* x    = (const float*)d_in[0];
    const float* ftw  = (const float*)d_in[1];
    const float* ftb  = (const float*)d_in[2];
    const float* rw1w = (const float*)d_in[3];
    const float* rw1b = (const float*)d_in[4];
    const float* bn1g = (const float*)d_in[5];
    const float* bn1b = (const float*)d_in[6];
    const float* rw2w = (const float*)d_in[7];
    const float* rw2b = (const float*)d_in[8];
    const float* nuw  = (const float*)d_in[9];
    const float* nub  = (const float*)d_in[10];
    const float* bn2g = (const float*)d_in[11];
    const float* bn2b = (const float*)d_in[12];
    float* out = (float*)d_out;

    // workspace carve-up (~37 MB)
    float* xt     = (float*)d_ws;            // 2097152
    float* xx     = xt + (size_t)BN * DD;    // 16384
    int*   nbr    = (int*)(xx + BN);         // 262144 ints
    float* wgt    = (float*)(nbr + NE);      // 262144
    float* agg    = wgt + NE;                // 2097152
    float* upre   = agg + (size_t)BN * DD;   // 2097152
    float* p1s    = upre + (size_t)BN * OO;  // 64 * 16384
    float* p1q    = p1s + (size_t)HH * 16384;
    float* p2s    = p1q + (size_t)HH * 16384;  // 128 * 1024
    float* p2q    = p2s + (size_t)OO * 1024;
    float* scale1 = p2q + (size_t)OO * 1024;   // 64
    float* shift1 = scale1 + HH;               // 64
    float* scale2 = shift1 + HH;               // 128
    float* shift2 = scale2 + OO;               // 128

    ft_gemm_kernel   <<<(BN / 16) * (DD / 16), 32, 0, stream>>>(x, ftw, ftb, xt);
    rownorm_kernel   <<<BN / 256, 256, 0, stream>>>(xt, xx);
    dist_topk_kernel <<<BB * (NN / 16), 32, 0, stream>>>(xt, xx, nbr);
    edge_stats_kernel<<<NE / 16, 32, 0, stream>>>(x, nbr, rw1w, rw1b, p1s, p1q);
    bn1_fin_kernel   <<<HH, 256, 0, stream>>>(p1s, p1q, bn1g, bn1b, rw1b, scale1, shift1);
    edge_weight_kernel<<<NE / 16, 32, 0, stream>>>(x, nbr, rw1w, scale1, shift1, rw2w, rw2b, wgt);
    aggregate_kernel <<<BN, DD, 0, stream>>>(x, nbr, wgt, agg);
    nu_gemm_kernel   <<<(BN / 16) * (OO / 16), 32, 0, stream>>>(x, agg, nuw, nub, upre, p2s, p2q);
    bn2_fin_kernel   <<<OO, 256, 0, stream>>>(p2s, p2q, bn2g, bn2b, scale2, shift2);
    out_kernel       <<<(BN * OO) / 256, 256, 0, stream>>>(upre, scale2, shift2, out);
}